// MeshGCN_80633716015488
// MI455X (gfx1250) — compile-verified
//
#include <hip/hip_runtime.h>

#define NN 100000
#define NE 600000
#define HID 128

typedef __attribute__((ext_vector_type(2))) float v2f;
typedef __attribute__((ext_vector_type(8))) float v8f;

// ---------------- degree / norm prep ----------------

__global__ void k_zero(float* __restrict__ p, int n) {
    int i = blockIdx.x * blockDim.x + threadIdx.x;
    if (i < n) p[i] = 0.0f;
}

__global__ void k_deg(float* __restrict__ deg, const int* __restrict__ dst) {
    int e = blockIdx.x * blockDim.x + threadIdx.x;
    if (e < NE) atomicAdd(&deg[dst[e]], 1.0f);
}

__global__ void k_dinv(float* __restrict__ deg) {
    int i = blockIdx.x * blockDim.x + threadIdx.x;
    if (i < NN) deg[i] = rsqrtf(deg[i] + 1.0f);
}

__global__ void k_enorm(float* __restrict__ enorm, const int* __restrict__ src,
                        const int* __restrict__ dst, const float* __restrict__ dinv) {
    int e = blockIdx.x * blockDim.x + threadIdx.x;
    if (e < NE) enorm[e] = dinv[src[e]] * dinv[dst[e]];
}

// ---------------- layer 0 GEMM: [N,6] @ [6,128] ----------------

__global__ void k_gemm_in6(const float* __restrict__ x, const float* __restrict__ sdf,
                           const float* __restrict__ W, float* __restrict__ Y) {
    int i = blockIdx.x * blockDim.x + threadIdx.x;
    if (i >= NN * HID) return;
    int node = i >> 7;
    int c = i & 127;
    const float* xr = x + node * 5;
    float s = xr[0] * W[0 * HID + c];
    s += xr[1] * W[1 * HID + c];
    s += xr[2] * W[2 * HID + c];
    s += xr[3] * W[3 * HID + c];
    s += xr[4] * W[4 * HID + c];
    s += sdf[node] * W[5 * HID + c];
    Y[i] = s;
}

// ---------------- hidden GEMM: relu(H) @ W, 128x128, fp32 WMMA ----------------
// One wave computes a 16-row x 128-col output slab using eight 16x16 tiles.
// A (16x4 fp32): per lane float2, low lanes hold K=k0..k0+1, high lanes K=k0+2..k0+3.
// B (4x16 fp32): same striping, N across lanes; staged transposed in LDS so the
// K-pair is a contiguous ds_load_b64.

__global__ __launch_bounds__(128) void k_gemm128_wmma(const float* __restrict__ H,
                                                      const float* __restrict__ W,
                                                      float* __restrict__ Y) {
    __shared__ float wt[HID * HID];  // wt[n*128 + k] = W[k*128 + n]  (64 KB)
    for (int i = threadIdx.x; i < HID * HID; i += 128) {
        int k = i >> 7;
        int n = i & 127;
        wt[n * HID + k] = W[i];
    }
    __syncthreads();

    int wave = threadIdx.x >> 5;
    int lane = threadIdx.x & 31;
    int half = lane >> 4;   // 0: lanes 0-15, 1: lanes 16-31
    int l15  = lane & 15;

    int mtile = blockIdx.x * 4 + wave;      // wave-uniform -> EXEC stays full
    if (mtile * 16 >= NN) return;
    int row_base = mtile * 16;

    const float* hrow = H + (size_t)(row_base + l15) * HID + 2 * half;

    v8f acc[8];
#pragma unroll
    for (int nt = 0; nt < 8; ++nt)
#pragma unroll
        for (int v = 0; v < 8; ++v) acc[nt][v] = 0.0f;

    for (int k0 = 0; k0 < HID; k0 += 4) {
        v2f a;
        a.x = fmaxf(hrow[k0 + 0], 0.0f);   // fused ReLU on input activations
        a.y = fmaxf(hrow[k0 + 1], 0.0f);
#pragma unroll
        for (int nt = 0; nt < 8; ++nt) {
            const float* wp = &wt[(nt * 16 + l15) * HID + k0 + 2 * half];
            v2f b;
            b.x = wp[0];
            b.y = wp[1];
            acc[nt] = __builtin_amdgcn_wmma_f32_16x16x4_f32(
                false, a, false, b, (short)0, acc[nt], false, false);
        }
    }

    // C/D layout: VGPR v -> M=v (lanes 0-15), M=v+8 (lanes 16-31); N = lane&15
#pragma unroll
    for (int nt = 0; nt < 8; ++nt) {
#pragma unroll
        for (int v = 0; v < 8; ++v) {
            int m = v + 8 * half;
            Y[(size_t)(row_base + m) * HID + nt * 16 + l15] = acc[nt][v];
        }
    }
}

// ---------------- last GEMM: relu(H) @ W5 -> [N,3] ----------------

__global__ void k_gemm_out3(const float* __restrict__ H, const float* __restrict__ W,
                            float* __restrict__ Y) {
    int node = blockIdx.x * blockDim.x + threadIdx.x;
    if (node >= NN) return;
    const float* h = H + (size_t)node * HID;
    float s0 = 0.f, s1 = 0.f, s2 = 0.f;
    for (int k = 0; k < HID; ++k) {
        float hv = fmaxf(h[k], 0.0f);
        s0 += hv * W[k * 3 + 0];
        s1 += hv * W[k * 3 + 1];
        s2 += hv * W[k * 3 + 2];
    }
    Y[node * 3 + 0] = s0;
    Y[node * 3 + 1] = s1;
    Y[node * 3 + 2] = s2;
}

// ---------------- aggregation: self term + bias, then edge scatter ----------------

__global__ void k_agg_init(float* __restrict__ out, const float* __restrict__ xw,
                           const float* __restrict__ dinv, const float* __restrict__ bias,
                           int d) {
    long long i = (long long)blockIdx.x * blockDim.x + threadIdx.x;
    if (i >= (long long)NN * d) return;
    int node = (int)(i / d);
    int c = (int)(i - (long long)node * d);
    float s = dinv[node];
    out[i] = xw[i] * s * s + bias[c];
}

__global__ void k_scatter128(float* __restrict__ out, const float* __restrict__ xw,
                             const int* __restrict__ src, const int* __restrict__ dst,
                             const float* __restrict__ enorm) {
    int t = blockIdx.x * blockDim.x + threadIdx.x;
    if (t >= NE * 32) return;
    int e  = t >> 5;
    int cg = (t & 31) << 2;
    int s  = src[e];
    int dd = dst[e];
    float nrm = enorm[e];
    const float4 v = *(const float4*)(xw + (size_t)s * HID + cg);
    float* op = out + (size_t)dd * HID + cg;
    atomicAdd(op + 0, nrm * v.x);
    atomicAdd(op + 1, nrm * v.y);
    atomicAdd(op + 2, nrm * v.z);
    atomicAdd(op + 3, nrm * v.w);
}

__global__ void k_scatter3(float* __restrict__ out, const float* __restrict__ xw,
                           const int* __restrict__ src, const int* __restrict__ dst,
                           const float* __restrict__ enorm) {
    int e = blockIdx.x * blockDim.x + threadIdx.x;
    if (e >= NE) return;
    int s  = src[e];
    int dd = dst[e];
    float nrm = enorm[e];
    atomicAdd(&out[dd * 3 + 0], nrm * xw[s * 3 + 0]);
    atomicAdd(&out[dd * 3 + 1], nrm * xw[s * 3 + 1]);
    atomicAdd(&out[dd * 3 + 2], nrm * xw[s * 3 + 2]);
}

// ---------------- launch ----------------

extern "C" void kernel_launch(void* const* d_in, const int* in_sizes, int n_in,
                              void* d_out, int out_size, void* d_ws, size_t ws_size,
                              hipStream_t stream) {
    const float* x   = (const float*)d_in[0];
    const float* sdf = (const float*)d_in[1];
    const int*   ei  = (const int*)d_in[2];
    const float* W[6];
    const float* B[6];
    for (int i = 0; i < 6; ++i) {
        W[i] = (const float*)d_in[3 + 2 * i];
        B[i] = (const float*)d_in[4 + 2 * i];
    }
    const int* src = ei;
    const int* dst = ei + NE;

    float* dinv  = (float*)d_ws;                 // NN floats
    float* enorm = dinv + NN;                    // NE floats
    float* bufA  = enorm + NE;                   // NN*128 floats
    float* bufB  = bufA + (size_t)NN * HID;      // NN*128 floats
    float* out   = (float*)d_out;

    const int T = 256;
    // degrees -> dinv -> per-edge norms
    k_zero <<<(NN + T - 1) / T, T, 0, stream>>>(dinv, NN);
    k_deg  <<<(NE + T - 1) / T, T, 0, stream>>>(dinv, dst);
    k_dinv <<<(NN + T - 1) / T, T, 0, stream>>>(dinv);
    k_enorm<<<(NE + T - 1) / T, T, 0, stream>>>(enorm, src, dst, dinv);

    // layer 0: concat(x,sdf) @ W0, then aggregate
    k_gemm_in6  <<<(NN * HID + T - 1) / T, T, 0, stream>>>(x, sdf, W[0], bufA);
    k_agg_init  <<<(NN * HID + T - 1) / T, T, 0, stream>>>(bufB, bufA, dinv, B[0], HID);
    k_scatter128<<<(NE * 32 + T - 1) / T, T, 0, stream>>>(bufB, bufA, src, dst, enorm);

    // layers 1..4: WMMA GEMM (ReLU fused on load) + aggregate
    const int mblocks = (NN / 16 + 3) / 4;   // 6250 M-tiles, 4 waves/block
    for (int l = 1; l <= 4; ++l) {
        k_gemm128_wmma<<<mblocks, 128, 0, stream>>>(bufB, W[l], bufA);
        k_agg_init    <<<(NN * HID + T - 1) / T, T, 0, stream>>>(bufB, bufA, dinv, B[l], HID);
        k_scatter128  <<<(NE * 32 + T - 1) / T, T, 0, stream>>>(bufB, bufA, src, dst, enorm);
    }

    // layer 5: relu(h) @ W5 -> [N,3], aggregate into d_out (no ReLU after)
    k_gemm_out3<<<(NN + T - 1) / T, T, 0, stream>>>(bufB, W[5], bufA);
    k_agg_init <<<(NN * 3 + T - 1) / T, T, 0, stream>>>(out, bufA, dinv, B[5], 3);
    k_scatter3 <<<(NE + T - 1) / T, T, 0, stream>>>(out, bufA, src, dst, enorm);
}